// _GATA_Official_11184094838824
// MI455X (gfx1250) — compile-verified
//
#include <hip/hip_runtime.h>
#include <math.h>

#define NN 10000
#define NE 100000
#define FDIM 128
#define HEADS 8

typedef float v2f __attribute__((ext_vector_type(2)));
typedef float v8f __attribute__((ext_vector_type(8)));
typedef unsigned int u32x4 __attribute__((ext_vector_type(4)));
typedef int i32x4 __attribute__((ext_vector_type(4)));
typedef int i32x8 __attribute__((ext_vector_type(8)));

// ---------------------------------------------------------------- helpers
__device__ __forceinline__ float blk_sum(float v, float* red) {
    int t = threadIdx.x;
    red[t] = v; __syncthreads();
    #pragma unroll
    for (int st = 64; st > 0; st >>= 1) { if (t < st) red[t] += red[t + st]; __syncthreads(); }
    float r = red[0]; __syncthreads();
    return r;
}
__device__ __forceinline__ float blk_max(float v, float* red) {
    int t = threadIdx.x;
    red[t] = v; __syncthreads();
    #pragma unroll
    for (int st = 64; st > 0; st >>= 1) { if (t < st) red[t] = fmaxf(red[t], red[t + st]); __syncthreads(); }
    float r = red[0]; __syncthreads();
    return r;
}
__device__ __forceinline__ float blk_min(float v, float* red) {
    int t = threadIdx.x;
    red[t] = v; __syncthreads();
    #pragma unroll
    for (int st = 64; st > 0; st >>= 1) { if (t < st) red[t] = fminf(red[t], red[t + st]); __syncthreads(); }
    float r = red[0]; __syncthreads();
    return r;
}
__device__ __forceinline__ void atomicMaxF(float* addr, float val) {
    int* ai = (int*)addr;
    int old = *ai;
    while (__int_as_float(old) < val) {
        int assumed = old;
        old = atomicCAS(ai, assumed, __float_as_int(val));
        if (old == assumed) break;
    }
}

// ---------------------------------------------------------------- 1. node prep
__global__ void k_node_prep(const float* __restrict__ s, const float* __restrict__ X,
                            const float* __restrict__ ln_w, const float* __restrict__ ln_b,
                            const float* __restrict__ tln_w,
                            float* __restrict__ s_ln, float* __restrict__ X_ln) {
    __shared__ float red[128];
    int n = blockIdx.x, t = threadIdx.x;
    float sv = s[(size_t)n * FDIM + t];
    float mu  = blk_sum(sv, red) * (1.0f / 128.0f);
    float dv  = sv - mu;
    float var = blk_sum(dv * dv, red) * (1.0f / 128.0f);
    s_ln[(size_t)n * FDIM + t] = dv * rsqrtf(var + 1e-5f) * ln_w[t] + ln_b[t];

    const float tw = tln_w[t];
    // group 0: l = 0..2
    {
        float p0 = X[(size_t)n * 1024 + 0 * 128 + t];
        float p1 = X[(size_t)n * 1024 + 1 * 128 + t];
        float p2 = X[(size_t)n * 1024 + 2 * 128 + t];
        float dist = fmaxf(sqrtf(p0 * p0 + p1 * p1 + p2 * p2), 1e-12f);
        float mx = blk_max(dist, red);
        float mn = blk_min(dist, red);
        float delta = mx - mn; delta = (delta == 0.f) ? 1.f : delta;
        float sc = fmaxf((dist - mn) / delta, 0.f) / dist * tw;
        X_ln[(size_t)n * 1024 + 0 * 128 + t] = p0 * sc;
        X_ln[(size_t)n * 1024 + 1 * 128 + t] = p1 * sc;
        X_ln[(size_t)n * 1024 + 2 * 128 + t] = p2 * sc;
    }
    // group 1: l = 3..7
    {
        float p[5]; float ss = 0.f;
        #pragma unroll
        for (int l = 0; l < 5; l++) { p[l] = X[(size_t)n * 1024 + (3 + l) * 128 + t]; ss += p[l] * p[l]; }
        float dist = fmaxf(sqrtf(ss), 1e-12f);
        float mx = blk_max(dist, red);
        float mn = blk_min(dist, red);
        float delta = mx - mn; delta = (delta == 0.f) ? 1.f : delta;
        float sc = fmaxf((dist - mn) / delta, 0.f) / dist * tw;
        #pragma unroll
        for (int l = 0; l < 5; l++) X_ln[(size_t)n * 1024 + (3 + l) * 128 + t] = p[l] * sc;
    }
}

// ---------------------------------------------------------------- 2. init accumulators
__global__ void k_init(const float* __restrict__ s_ln, const float* __restrict__ X_ln,
                       float* __restrict__ out_s, float* __restrict__ out_X,
                       float* __restrict__ nmax, float* __restrict__ nden, float* __restrict__ deg) {
    size_t i = (size_t)blockIdx.x * blockDim.x + threadIdx.x;
    if (i < (size_t)NN * 1024) out_X[i] = X_ln[i];
    if (i < (size_t)NN * 128)  out_s[i] = s_ln[i];
    if (i < (size_t)NN * HEADS) { nmax[i] = -3.0e38f; nden[i] = 0.f; }
    if (i < (size_t)NN) deg[i] = 0.f;
}

// ---------------------------------------------------------------- 3. generic fp32 WMMA GEMM
// C[M,N] = act(A[M,128] @ W[N,128]^T + bias) * cutoff(dvec[row])
// grid = (M/16, Ntiles/4), block = 128 (4 waves; wave w -> N-tile blockIdx.y*4+w)
// A tile (16x128 fp32, 8KB) staged to LDS by the Tensor Data Mover (one wave
// issues the D#-described DMA, waits on TENSORcnt, block barrier publishes it).
__global__ void k_gemm_wmma(const float* __restrict__ A, const float* __restrict__ W,
                            const float* __restrict__ bias, float* __restrict__ C,
                            int M, int N, int act, const float* __restrict__ dvec) {
    __shared__ float lA[16 * 128];
    int mt = blockIdx.x;
    int lane = threadIdx.x & 31;
    int wv = threadIdx.x >> 5;

#if defined(__HIP_DEVICE_COMPILE__) && __has_builtin(__builtin_amdgcn_tensor_load_to_lds)
    if (wv == 0) {
        unsigned long long gaddr = (unsigned long long)(A + (size_t)mt * 16 * 128);
        u32x4 g0;
        g0.x = 1u;                                                      // count=1, user mode
        g0.y = (unsigned)(size_t)(&lA[0]);                              // lds_addr
        g0.z = (unsigned)(gaddr & 0xFFFFFFFFu);                         // global_addr[31:0]
        g0.w = (unsigned)((gaddr >> 32) & 0x01FFFFFFu) | 0x80000000u;   // addr[56:32] | type=2
        i32x8 g1;
        g1[0] = 0x00020000;                                   // data_size=2 (4B), no mask/pad
        g1[1] = (int)(128u << 16);                            // tensor_dim0[15:0]=128
        g1[2] = (int)(((unsigned)M & 0xFFFFu) << 16);         // dim0 hi=0 | tensor_dim1[15:0]
        g1[3] = (int)(((unsigned)M >> 16) | (128u << 16));    // dim1 hi | tile_dim0=128
        g1[4] = 16;                                           // tile_dim1=16, tile_dim2=0
        g1[5] = 128;                                          // tensor_dim0_stride=128
        g1[6] = 0;
        g1[7] = 0;
        i32x4 z4 = {0, 0, 0, 0};
        i32x8 z8 = {0, 0, 0, 0, 0, 0, 0, 0};
        __builtin_amdgcn_tensor_load_to_lds(g0, g1, z4, z4, z8, 0);
        __builtin_amdgcn_s_wait_tensorcnt(0);
    }
#else
    for (int i = threadIdx.x; i < 16 * 128; i += 128) {
        int r = i >> 7, c = i & 127;
        lA[i] = A[(size_t)(mt * 16 + r) * 128 + c];
    }
#endif
    __syncthreads();

    int nt  = blockIdx.y * 4 + wv;
    int row  = lane & 15;
    int koff = (lane >> 4) * 2;
    int col  = nt * 16 + (lane & 15);
    const float* wcol = W + (size_t)col * 128;
    v8f acc = {};
    #pragma unroll
    for (int k0 = 0; k0 < 128; k0 += 4) {
        v2f a, b;
        a.x = lA[row * 128 + k0 + koff];
        a.y = lA[row * 128 + k0 + koff + 1];
        b.x = wcol[k0 + koff];
        b.y = wcol[k0 + koff + 1];
        acc = __builtin_amdgcn_wmma_f32_16x16x4_f32(false, a, false, b, (short)0, acc, false, false);
    }
    float bcol = bias[col];
    #pragma unroll
    for (int v = 0; v < 8; v++) {
        int m = mt * 16 + v + ((lane >> 4) * 8);
        float val = acc[v] + bcol;
        if (act == 1) val = val / (1.f + __expf(-val));          // silu
        if (dvec) {
            float d = dvec[m];
            float cut = (d < 5.0f) ? 0.5f * (__cosf(d * 0.62831853071f) + 1.f) : 0.f;
            val *= cut;
        }
        C[(size_t)m * N + col] = val;
    }
}

// ---------------------------------------------------------------- 4a. raw attention + segment max + degree
__global__ void k_attn_raw(const float* __restrict__ q, const float* __restrict__ k,
                           const float* __restrict__ t_attn,
                           const int* __restrict__ ecv, const int* __restrict__ env,
                           float* __restrict__ raw, float* __restrict__ nmax, float* __restrict__ deg) {
    int gid = blockIdx.x * blockDim.x + threadIdx.x;
    if (gid >= NE * HEADS) return;
    int e = gid >> 3, h = gid & 7;
    int c = ecv[e], n = env[e];
    const float* qp = q + (size_t)c * 128 + h * 16;
    const float* kp = k + (size_t)n * 128 + h * 16;
    const float* tp = t_attn + (size_t)e * 128 + h * 16;
    float dot = 0.f;
    #pragma unroll
    for (int d = 0; d < 16; d++) dot += qp[d] * kp[d] * tp[d];
    raw[gid] = dot;
    atomicMaxF(&nmax[c * 8 + h], dot);
    if (h == 0) atomicAdd(&deg[c], 1.0f);
}

// ---------------------------------------------------------------- 4b. exp + segment sum
__global__ void k_attn_exp(const int* __restrict__ ecv, float* __restrict__ raw,
                           const float* __restrict__ nmax, float* __restrict__ nden) {
    int gid = blockIdx.x * blockDim.x + threadIdx.x;
    if (gid >= NE * HEADS) return;
    int e = gid >> 3, h = gid & 7;
    int c = ecv[e];
    float v = __expf(raw[gid] - nmax[c * 8 + h]);
    raw[gid] = v;
    atomicAdd(&nden[c * 8 + h], v);
}

// ---------------------------------------------------------------- 5. attend + scatter to s_out / X_out
__global__ void k_attend_scatter(const int* __restrict__ ecv, const int* __restrict__ env,
                                 const float* __restrict__ xg, const float* __restrict__ vg,
                                 const float* __restrict__ t_fil, const float* __restrict__ ea,
                                 const float* __restrict__ nden, const float* __restrict__ deg,
                                 const float* __restrict__ rl, const float* __restrict__ X_ln,
                                 float* __restrict__ s_out, float* __restrict__ X_out) {
    int e = blockIdx.x, t = threadIdx.x;
    int c = ecv[e], n = env[e];
    float degs = sqrtf(deg[c]) * 0.08838834764831845f;   // 1/sqrt(128)
    int hs = t / 48, hd = (128 + t) / 48, ht = (256 + t) / 48;
    float cs = ea[e * 8 + hs] / nden[c * 8 + hs] * degs;
    float cd = ea[e * 8 + hd] / nden[c * 8 + hd] * degs;
    float ct = ea[e * 8 + ht] / nden[c * 8 + ht] * degs;
    float os = xg[(size_t)n * 384 + t]       * t_fil[(size_t)e * 384 + t]       * cs;
    float od = vg[(size_t)n * 384 + 128 + t] * t_fil[(size_t)e * 384 + 128 + t] * cd;
    float ot = vg[(size_t)n * 384 + 256 + t] * t_fil[(size_t)e * 384 + 256 + t] * ct;
    atomicAdd(&s_out[(size_t)c * 128 + t], os);
    #pragma unroll
    for (int l = 0; l < 8; l++) {
        float dx = od * rl[e * 8 + l] + ot * X_ln[(size_t)n * 1024 + l * 128 + t];
        atomicAdd(&X_out[(size_t)c * 1024 + l * 128 + t], dx);
    }
}

// ---------------------------------------------------------------- 6. per-edge w1/w2 WMMA + gate
// block = 64 (2 waves). Each wave handles an edge PAIR: A rows 0..7 = edge0 l, 8..15 = edge1 l.
// Gathered X_out rows are staged to LDS with global_load_async_to_lds_b128 (ASYNCcnt).
__global__ void k_edge_gate(const int* __restrict__ ecv, const int* __restrict__ env,
                            const float* __restrict__ X_out, const float* __restrict__ Wvq,
                            const float* __restrict__ Wvk, const float* __restrict__ rl,
                            const float* __restrict__ f_ij, const float* __restrict__ gedge,
                            float* __restrict__ f_out) {
    __shared__ float lds[2 * 2 * 16 * 128];   // per wave: A1 (X_out[ec]) + A2 (X_out[en])
    int wv = threadIdx.x >> 5, lane = threadIdx.x & 31;
    int pair = blockIdx.x * 2 + wv;
    int e0 = pair * 2, e1 = pair * 2 + 1;
    float* A1 = lds + wv * 4096;
    float* A2 = A1 + 2048;
    int c0 = ecv[e0], c1 = ecv[e1], n0 = env[e0], n1 = env[e1];

#if defined(__HIP_DEVICE_COMPILE__) && defined(__gfx1250__)
    {
        unsigned l1 = (unsigned)(size_t)(&A1[0]);
        unsigned l2 = (unsigned)(size_t)(&A2[0]);
        #pragma unroll
        for (int it = 0; it < 16; ++it) {
            int chunk = lane + it * 32;           // 16-byte chunk id (0..511)
            int elem  = chunk * 4;
            int r = elem >> 7, cc = elem & 127;
            int nc = (r < 8) ? c0 : c1;
            int nn = (r < 8) ? n0 : n1;
            int l  = r & 7;
            const float* g1p = X_out + (size_t)nc * 1024 + l * 128 + cc;
            const float* g2p = X_out + (size_t)nn * 1024 + l * 128 + cc;
            unsigned d1 = l1 + (unsigned)chunk * 16u;
            unsigned d2 = l2 + (unsigned)chunk * 16u;
            asm volatile("global_load_async_to_lds_b128 %0, %1, off"
                         :: "v"(d1), "v"(g1p) : "memory");
            asm volatile("global_load_async_to_lds_b128 %0, %1, off"
                         :: "v"(d2), "v"(g2p) : "memory");
        }
        asm volatile("s_wait_asynccnt 0x0" ::: "memory");
    }
#else
    for (int i = lane; i < 2048; i += 32) {
        int r = i >> 7, c = i & 127;
        int ee = (r < 8) ? e0 : e1;
        int l  = r & 7;
        A1[i] = X_out[(size_t)ecv[ee] * 1024 + l * 128 + c];
        A2[i] = X_out[(size_t)env[ee] * 1024 + l * 128 + c];
    }
#endif
    __syncthreads();

    int e = (lane < 16) ? e0 : e1;
    float rlr[8];
    #pragma unroll
    for (int l = 0; l < 8; l++) rlr[l] = rl[e * 8 + l];
    int row  = lane & 15;
    int koff = (lane >> 4) * 2;
    const float* Wvk1 = Wvk + 128 * 128;

    for (int nf = 0; nf < 8; nf++) {
        int col = nf * 16 + (lane & 15);
        const float* w1c  = Wvq  + (size_t)col * 128;
        const float* w2ac = Wvk  + (size_t)col * 128;
        const float* w2bc = Wvk1 + (size_t)col * 128;
        v8f acc1 = {}, acc2a = {}, acc2b = {};
        #pragma unroll 8
        for (int k0 = 0; k0 < 128; k0 += 4) {
            v2f a1, a2, b1, b2a, b2b;
            a1.x = A1[row * 128 + k0 + koff]; a1.y = A1[row * 128 + k0 + koff + 1];
            a2.x = A2[row * 128 + k0 + koff]; a2.y = A2[row * 128 + k0 + koff + 1];
            b1.x  = w1c[k0 + koff];  b1.y  = w1c[k0 + koff + 1];
            b2a.x = w2ac[k0 + koff]; b2a.y = w2ac[k0 + koff + 1];
            b2b.x = w2bc[k0 + koff]; b2b.y = w2bc[k0 + koff + 1];
            acc1  = __builtin_amdgcn_wmma_f32_16x16x4_f32(false, a1, false, b1,  (short)0, acc1,  false, false);
            acc2a = __builtin_amdgcn_wmma_f32_16x16x4_f32(false, a2, false, b2a, (short)0, acc2a, false, false);
            acc2b = __builtin_amdgcn_wmma_f32_16x16x4_f32(false, a2, false, b2b, (short)0, acc2b, false, false);
        }
        // acc VGPR v holds l = v of this lane's edge (lanes<16: edge0, lanes>=16: edge1)
        float w1v[8], w2v[8], s1 = 0.f, s2 = 0.f;
        #pragma unroll
        for (int v = 0; v < 8; v++) {
            w1v[v] = acc1[v];
            w2v[v] = (v < 3) ? acc2a[v] : acc2b[v];   // group0 channels use Wvk[0]
            s1 += w1v[v] * rlr[v];
            s2 += w2v[v] * rlr[v];                    // (-rl)·(-rl) sign cancels
        }
        float wdot = 0.f;
        #pragma unroll
        for (int v = 0; v < 8; v++) {
            float a = w1v[v] - s1 * rlr[v];
            float b = w2v[v] - s2 * rlr[v];
            wdot += a * b;
        }
        size_t idx = (size_t)e * 128 + col;
        f_out[idx] = f_ij[idx] + gedge[idx] * wdot;
    }
}

// ---------------------------------------------------------------- launch
extern "C" void kernel_launch(void* const* d_in, const int* in_sizes, int n_in,
                              void* d_out, int out_size, void* d_ws, size_t ws_size,
                              hipStream_t stream) {
    const float* s     = (const float*)d_in[0];
    const float* X     = (const float*)d_in[1];
    const float* f_ij  = (const float*)d_in[2];
    const float* rl_ij = (const float*)d_in[3];
    const float* d_ij  = (const float*)d_in[4];
    const int*   ec    = (const int*)d_in[5];
    const int*   en    = (const int*)d_in[6];
    const float* ln_w  = (const float*)d_in[7];
    const float* ln_b  = (const float*)d_in[8];
    const float* tln_w = (const float*)d_in[9];
    const float* Wq_w  = (const float*)d_in[10];
    const float* Wq_b  = (const float*)d_in[11];
    const float* Wk_w  = (const float*)d_in[12];
    const float* Wk_b  = (const float*)d_in[13];
    const float* gs1_w = (const float*)d_in[14];
    const float* gs1_b = (const float*)d_in[15];
    const float* gs2_w = (const float*)d_in[16];
    const float* gs2_b = (const float*)d_in[17];
    const float* gv1_w = (const float*)d_in[18];
    const float* gv1_b = (const float*)d_in[19];
    const float* gv2_w = (const float*)d_in[20];
    const float* gv2_b = (const float*)d_in[21];
    const float* Wre_w = (const float*)d_in[22];
    const float* Wre_b = (const float*)d_in[23];
    const float* Wrs_w = (const float*)d_in[24];
    const float* Wrs_b = (const float*)d_in[25];
    const float* gt1_w = (const float*)d_in[26];
    const float* gt1_b = (const float*)d_in[27];
    const float* Wvq_w = (const float*)d_in[28];
    const float* Wvk_w = (const float*)d_in[29];

    float* ws = (float*)d_ws;
    size_t o = 0;
    float* s_ln  = ws + o; o += (size_t)NN * 128;
    float* X_ln  = ws + o; o += (size_t)NN * 1024;
    float* qb    = ws + o; o += (size_t)NN * 128;
    float* kb    = ws + o; o += (size_t)NN * 128;
    float* hx    = ws + o; o += (size_t)NN * 128;
    float* xg    = ws + o; o += (size_t)NN * 384;
    float* hv    = ws + o; o += (size_t)NN * 128;
    float* vg    = ws + o; o += (size_t)NN * 384;
    float* tatt  = ws + o; o += (size_t)NE * 128;
    float* tfil  = ws + o; o += (size_t)NE * 384;
    float* gedge = ws + o; o += (size_t)NE * 128;
    float* eaw   = ws + o; o += (size_t)NE * HEADS;
    float* nmax  = ws + o; o += (size_t)NN * HEADS;
    float* nden  = ws + o; o += (size_t)NN * HEADS;
    float* deg   = ws + o; o += (size_t)NN;

    float* out_s = (float*)d_out;
    float* out_X = out_s + (size_t)NN * 128;
    float* out_f = out_X + (size_t)NN * 1024;

    // 1. node prep
    k_node_prep<<<NN, 128, 0, stream>>>(s, X, ln_w, ln_b, tln_w, s_ln, X_ln);
    // 2. init outputs + softmax accumulators
    k_init<<<(NN * 1024 + 255) / 256, 256, 0, stream>>>(s_ln, X_ln, out_s, out_X, nmax, nden, deg);
    // 3. GEMMs (fp32 WMMA, TDM-staged A tiles): node side
    k_gemm_wmma<<<dim3(625, 2), 128, 0, stream>>>(s_ln, Wq_w, Wq_b, qb, NN, 128, 0, nullptr);
    k_gemm_wmma<<<dim3(625, 2), 128, 0, stream>>>(s_ln, Wk_w, Wk_b, kb, NN, 128, 0, nullptr);
    k_gemm_wmma<<<dim3(625, 2), 128, 0, stream>>>(s_ln, gs1_w, gs1_b, hx, NN, 128, 1, nullptr);
    k_gemm_wmma<<<dim3(625, 6), 128, 0, stream>>>(hx, gs2_w, gs2_b, xg, NN, 384, 0, nullptr);
    k_gemm_wmma<<<dim3(625, 2), 128, 0, stream>>>(s_ln, gv1_w, gv1_b, hv, NN, 128, 1, nullptr);
    k_gemm_wmma<<<dim3(625, 6), 128, 0, stream>>>(hv, gv2_w, gv2_b, vg, NN, 384, 0, nullptr);
    // edge side
    k_gemm_wmma<<<dim3(6250, 2), 128, 0, stream>>>(f_ij, Wre_w, Wre_b, tatt, NE, 128, 0, nullptr);
    k_gemm_wmma<<<dim3(6250, 6), 128, 0, stream>>>(f_ij, Wrs_w, Wrs_b, tfil, NE, 384, 0, d_ij);
    k_gemm_wmma<<<dim3(6250, 2), 128, 0, stream>>>(f_ij, gt1_w, gt1_b, gedge, NE, 128, 0, nullptr);
    // 4. segment softmax
    k_attn_raw<<<(NE * HEADS + 255) / 256, 256, 0, stream>>>(qb, kb, tatt, ec, en, eaw, nmax, deg);
    k_attn_exp<<<(NE * HEADS + 255) / 256, 256, 0, stream>>>(ec, eaw, nmax, nden);
    // 5. attend + scatter
    k_attend_scatter<<<NE, 128, 0, stream>>>(ec, en, xg, vg, tfil, eaw, nden, deg, rl_ij, X_ln, out_s, out_X);
    // 6. per-edge gate (WMMA + async-LDS gathers, 2 edges per wave, 2 waves per block)
    k_edge_gate<<<NE / 4, 64, 0, stream>>>(ec, en, out_X, Wvq_w, Wvk_w, rl_ij, f_ij, gedge, out_f);
}